// PortfolioGAT_63324997812963
// MI455X (gfx1250) — compile-verified
//
#include <hip/hip_runtime.h>
#include <hip/hip_bf16.h>

#define NN      100000                 // nodes
#define NE      1600000                // raw edges
#define ET      (NE + NN)              // edges incl. self loops
#define IN_F    128
#define HID     16
#define OUT1    32                     // 2 heads * 16
#define NEG_SLOPE 0.2f
#define NTILES  (NN / 16)              // 6250, exact

typedef float v2f __attribute__((ext_vector_type(2)));
typedef float v8f __attribute__((ext_vector_type(8)));

// ---------------- helpers ----------------
__device__ __forceinline__ void edge_nodes(const int* __restrict__ ei, int e, int& s, int& d) {
  if (e < NE) { s = ei[e]; d = ei[NE + e]; }
  else        { s = e - NE; d = e - NE; }          // appended self loops
}
__device__ __forceinline__ float lrelu(float v) { return v > 0.f ? v : NEG_SLOPE * v; }
// monotone float -> uint key for atomicMax on signed floats
__device__ __forceinline__ unsigned fkey(float f) {
  unsigned b = __float_as_uint(f);
  return (b & 0x80000000u) ? ~b : (b | 0x80000000u);
}
__device__ __forceinline__ float kfloat(unsigned k) {
  return (k & 0x80000000u) ? __uint_as_float(k & 0x7fffffffu) : __uint_as_float(~k);
}

// ---------------- Kernel 1: h1 = x @ W1 via V_WMMA_F32_16X16X4_F32 ----------------
// One wave per 16-row tile, two 16x16 column tiles accumulated over K=128 (32 WMMA steps x 2).
__launch_bounds__(256)
__global__ void gat_gemm1_wmma(const float* __restrict__ x, const float* __restrict__ W1,
                               float* __restrict__ h1) {
  __shared__ float w[IN_F * OUT1];                 // 16 KB of 320 KB WGP LDS
  for (int i = threadIdx.x; i < IN_F * OUT1; i += 256) w[i] = W1[i];
  __syncthreads();

  const int wave = threadIdx.x >> 5;
  const int lane = threadIdx.x & 31;
  const int tile = blockIdx.x * 8 + wave;
  if (tile >= NTILES) return;

  const int m  = lane & 15;                        // row within tile (A) / col within N-tile (B,C)
  const int kh = lane >> 4;                        // K sub-slot selector per ISA A/B layout
  const float* __restrict__ xrow = x + (size_t)(tile * 16 + m) * IN_F + 2 * kh;

  v8f c0 = {}; v8f c1 = {};
#pragma unroll 4
  for (int k = 0; k < IN_F; k += 4) {
    v2f a = *(const v2f*)(xrow + k);               // A[M=m][k+2kh], A[M=m][k+2kh+1]
    const int kk = k + 2 * kh;
    v2f b0, b1;
    b0.x = w[kk * OUT1 + m];        b0.y = w[(kk + 1) * OUT1 + m];        // N = 0..15
    b1.x = w[kk * OUT1 + 16 + m];   b1.y = w[(kk + 1) * OUT1 + 16 + m];  // N = 16..31
    c0 = __builtin_amdgcn_wmma_f32_16x16x4_f32(false, a, false, b0, (short)0, c0, false, false);
    c1 = __builtin_amdgcn_wmma_f32_16x16x4_f32(false, a, false, b1, (short)0, c1, false, false);
  }

#pragma unroll
  for (int r = 0; r < 8; ++r) {                    // C[M = r + 8*kh][N = m]
    const int row = tile * 16 + r + 8 * kh;
    h1[(size_t)row * OUT1 + m]       = c0[r];
    h1[(size_t)row * OUT1 + 16 + m]  = c1[r];
  }
}

// ---------------- Kernel 2: per-node scores for layer 1 + state init ----------------
__global__ void gat_node_prep1(const float* __restrict__ h1,
                               const float* __restrict__ a_src1, const float* __restrict__ a_dst1,
                               float* __restrict__ as1, float* __restrict__ ad1,
                               unsigned* __restrict__ maxk1, float* __restrict__ den1,
                               float* __restrict__ agg1) {
  const int n = blockIdx.x * 256 + threadIdx.x;
  if (n >= NN) return;
  const float* hr = h1 + (size_t)n * OUT1;
#pragma unroll
  for (int hd = 0; hd < 2; ++hd) {
    float s = 0.f, d = 0.f;
#pragma unroll
    for (int c = 0; c < HID; ++c) {
      const float v = hr[hd * HID + c];
      s = fmaf(v, a_src1[hd * HID + c], s);
      d = fmaf(v, a_dst1[hd * HID + c], d);
    }
    as1[n * 2 + hd] = s;
    ad1[n * 2 + hd] = d;
    maxk1[n * 2 + hd] = 0u;                        // key of -max float (ordered-uint space)
    den1[n * 2 + hd] = 0.f;
  }
  float* ag = agg1 + (size_t)n * OUT1;
#pragma unroll
  for (int c = 0; c < OUT1; ++c) ag[c] = 0.f;
}

// ---------------- Layer-1 edge passes ----------------
__global__ void gat_edge_max1(const int* __restrict__ ei, const float* __restrict__ as1,
                              const float* __restrict__ ad1, unsigned* __restrict__ maxk1) {
  const int e = blockIdx.x * 256 + threadIdx.x;
  if (e >= ET) return;
  int s, d; edge_nodes(ei, e, s, d);
#pragma unroll
  for (int hd = 0; hd < 2; ++hd) {
    const float sc = lrelu(as1[s * 2 + hd] + ad1[d * 2 + hd]);
    atomicMax(&maxk1[d * 2 + hd], fkey(sc));
  }
}

__global__ void gat_edge_den1(const int* __restrict__ ei, const float* __restrict__ as1,
                              const float* __restrict__ ad1, const unsigned* __restrict__ maxk1,
                              float* __restrict__ den1) {
  const int e = blockIdx.x * 256 + threadIdx.x;
  if (e >= ET) return;
  int s, d; edge_nodes(ei, e, s, d);
#pragma unroll
  for (int hd = 0; hd < 2; ++hd) {
    const float sc = lrelu(as1[s * 2 + hd] + ad1[d * 2 + hd]);
    atomicAdd(&den1[d * 2 + hd], __expf(sc - kfloat(maxk1[d * 2 + hd])));
  }
}

__global__ void gat_edge_agg1(const int* __restrict__ ei, const float* __restrict__ as1,
                              const float* __restrict__ ad1, const unsigned* __restrict__ maxk1,
                              const float* __restrict__ den1, const float* __restrict__ h1,
                              float* __restrict__ agg1) {
  const int e = blockIdx.x * 256 + threadIdx.x;
  if (e >= ET) return;
  int s, d; edge_nodes(ei, e, s, d);
  float w01[2];
#pragma unroll
  for (int hd = 0; hd < 2; ++hd) {
    const float sc = lrelu(as1[s * 2 + hd] + ad1[d * 2 + hd]);
    w01[hd] = __expf(sc - kfloat(maxk1[d * 2 + hd])) / (den1[d * 2 + hd] + 1e-16f);
  }
  const float* __restrict__ hs = h1 + (size_t)s * OUT1;   // L2-resident gather (12.8 MB)
  float* __restrict__ ag = agg1 + (size_t)d * OUT1;       // L2-resident scatter
#pragma unroll
  for (int c = 0; c < HID; ++c)      atomicAdd(&ag[c], w01[0] * hs[c]);
#pragma unroll
  for (int c = HID; c < OUT1; ++c)   atomicAdd(&ag[c], w01[1] * hs[c]);
}

// ---------------- Kernel: bias + ELU + layer-2 projection + state init ----------------
__global__ void gat_finish1(const float* __restrict__ agg1, const float* __restrict__ b1,
                            const float* __restrict__ W2,
                            float* __restrict__ h2, unsigned* __restrict__ maxk2,
                            float* __restrict__ den2, float* __restrict__ agg2) {
  const int n = blockIdx.x * 256 + threadIdx.x;
  if (n >= NN) return;
  const float* ag = agg1 + (size_t)n * OUT1;
  float acc = 0.f;
#pragma unroll
  for (int c = 0; c < OUT1; ++c) {
    float v = ag[c] + b1[c];
    v = v > 0.f ? v : (__expf(v) - 1.f);            // ELU
    acc = fmaf(v, W2[c], acc);                      // x2 @ W2 (32 -> 1)
  }
  h2[n] = acc;
  maxk2[n] = 0u;
  den2[n] = 0.f;
  agg2[n] = 0.f;
}

// ---------------- Layer-2 edge passes (1 head, 1 channel) ----------------
__global__ void gat_edge_max2(const int* __restrict__ ei, const float* __restrict__ h2,
                              const float* __restrict__ a_src2, const float* __restrict__ a_dst2,
                              unsigned* __restrict__ maxk2) {
  const int e = blockIdx.x * 256 + threadIdx.x;
  if (e >= ET) return;
  int s, d; edge_nodes(ei, e, s, d);
  const float sc = lrelu(h2[s] * a_src2[0] + h2[d] * a_dst2[0]);
  atomicMax(&maxk2[d], fkey(sc));
}

__global__ void gat_edge_den2(const int* __restrict__ ei, const float* __restrict__ h2,
                              const float* __restrict__ a_src2, const float* __restrict__ a_dst2,
                              const unsigned* __restrict__ maxk2, float* __restrict__ den2) {
  const int e = blockIdx.x * 256 + threadIdx.x;
  if (e >= ET) return;
  int s, d; edge_nodes(ei, e, s, d);
  const float sc = lrelu(h2[s] * a_src2[0] + h2[d] * a_dst2[0]);
  atomicAdd(&den2[d], __expf(sc - kfloat(maxk2[d])));
}

__global__ void gat_edge_agg2(const int* __restrict__ ei, const float* __restrict__ h2,
                              const float* __restrict__ a_src2, const float* __restrict__ a_dst2,
                              const unsigned* __restrict__ maxk2, const float* __restrict__ den2,
                              float* __restrict__ agg2) {
  const int e = blockIdx.x * 256 + threadIdx.x;
  if (e >= ET) return;
  int s, d; edge_nodes(ei, e, s, d);
  const float sc = lrelu(h2[s] * a_src2[0] + h2[d] * a_dst2[0]);
  const float w  = __expf(sc - kfloat(maxk2[d])) / (den2[d] + 1e-16f);
  atomicAdd(&agg2[d], w * h2[s]);
}

__global__ void gat_final(const float* __restrict__ agg2, const float* __restrict__ b2,
                          float* __restrict__ out) {
  const int n = blockIdx.x * 256 + threadIdx.x;
  if (n >= NN) return;
  out[n] = agg2[n] + b2[0];                         // heads=1 mean == identity
}

// ---------------- launch ----------------
extern "C" void kernel_launch(void* const* d_in, const int* in_sizes, int n_in,
                              void* d_out, int out_size, void* d_ws, size_t ws_size,
                              hipStream_t stream) {
  const float* x      = (const float*)d_in[0];
  const int*   ei     = (const int*)  d_in[1];
  const float* W1     = (const float*)d_in[2];
  const float* a_src1 = (const float*)d_in[3];
  const float* a_dst1 = (const float*)d_in[4];
  const float* b1     = (const float*)d_in[5];
  const float* W2     = (const float*)d_in[6];
  const float* a_src2 = (const float*)d_in[7];
  const float* a_dst2 = (const float*)d_in[8];
  const float* b2     = (const float*)d_in[9];
  float* out = (float*)d_out;

  // workspace carve-up (floats); total ~7.6M floats = 30.4 MB
  float* ws = (float*)d_ws;
  float*    h1    = ws;                              // 3.2M
  float*    as1   = h1   + (size_t)NN * OUT1;        // 0.2M
  float*    ad1   = as1  + (size_t)NN * 2;           // 0.2M
  unsigned* maxk1 = (unsigned*)(ad1 + (size_t)NN*2); // 0.2M
  float*    den1  = (float*)(maxk1 + (size_t)NN*2);  // 0.2M
  float*    agg1  = den1 + (size_t)NN * 2;           // 3.2M
  float*    h2    = agg1 + (size_t)NN * OUT1;        // 0.1M
  unsigned* maxk2 = (unsigned*)(h2 + NN);            // 0.1M
  float*    den2  = (float*)(maxk2 + NN);            // 0.1M
  float*    agg2  = den2 + NN;                       // 0.1M

  const int nodeBlocks = (NN + 255) / 256;           // 391
  const int edgeBlocks = (ET + 255) / 256;           // 6641
  const int gemmBlocks = (NTILES + 7) / 8;           // 782

  gat_gemm1_wmma<<<gemmBlocks, 256, 0, stream>>>(x, W1, h1);
  gat_node_prep1<<<nodeBlocks, 256, 0, stream>>>(h1, a_src1, a_dst1, as1, ad1, maxk1, den1, agg1);
  gat_edge_max1<<<edgeBlocks, 256, 0, stream>>>(ei, as1, ad1, maxk1);
  gat_edge_den1<<<edgeBlocks, 256, 0, stream>>>(ei, as1, ad1, maxk1, den1);
  gat_edge_agg1<<<edgeBlocks, 256, 0, stream>>>(ei, as1, ad1, maxk1, den1, h1, agg1);
  gat_finish1<<<nodeBlocks, 256, 0, stream>>>(agg1, b1, W2, h2, maxk2, den2, agg2);
  gat_edge_max2<<<edgeBlocks, 256, 0, stream>>>(ei, h2, a_src2, a_dst2, maxk2);
  gat_edge_den2<<<edgeBlocks, 256, 0, stream>>>(ei, h2, a_src2, a_dst2, maxk2, den2);
  gat_edge_agg2<<<edgeBlocks, 256, 0, stream>>>(ei, h2, a_src2, a_dst2, maxk2, den2, agg2);
  gat_final<<<nodeBlocks, 256, 0, stream>>>(agg2, b2, out);
}